// P2MLoss_32298154066350
// MI455X (gfx1250) — compile-verified
//
#include <hip/hip_runtime.h>
#include <hip/hip_bf16.h>
#include <math.h>

typedef float v2f __attribute__((ext_vector_type(2)));
typedef float v8f __attribute__((ext_vector_type(8)));

#define PGT 10000
#define NB 4

__device__ __forceinline__ v8f wmma_dist(v2f a, v2f b, v8f c) {
    // D = A(16x4,f32) x B(4x16,f32) + C(16x16,f32)
    return __builtin_amdgcn_wmma_f32_16x16x4_f32(
        /*neg_a=*/false, a, /*neg_b=*/false, b,
        /*c_mod=*/(short)0, c, /*reuse_a=*/false, /*reuse_b=*/false);
}

// Single-instruction float min: avoids the v_max_num canonicalization pair
// that strict fminf lowering inserts around v_min_num_f32.
__device__ __forceinline__ float fmin1(float a, float b) {
    float d;
    asm("v_min_num_f32 %0, %1, %2" : "=v"(d) : "v"(a), "v"(b));
    return d;
}

// ---------------------------------------------------------------------------
// Zero the accumulator slots in workspace
// ---------------------------------------------------------------------------
__global__ void zero_acc(float* acc) {
    if (threadIdx.x < 32) acc[threadIdx.x] = 0.f;
}

// ---------------------------------------------------------------------------
// Chamfer pred->gt: one wave per (pred strip of 16, batch).
// WMMA computes the partial tile  s[m][q] = |y_q|^2 - 2 x_m . y_q  with C=0
// (inline constant); |x_m|^2 is row-constant so it's added after the min
// reduction (min/argmin are invariant to it).
// A row m = (x0,x1,x2,1); B col q = (-2y0,-2y1,-2y2,|y|^2).
// Batch 0 additionally tracks argmin (needed for the normal loss).
// ---------------------------------------------------------------------------
__global__ __launch_bounds__(32) void chamfer_pg(
        const float* __restrict__ pred, const float* __restrict__ gt,
        int P, float* __restrict__ accum, int* __restrict__ idx0) {
    __shared__ float x2s[16];
    __shared__ float vmin[32][8];
    __shared__ int   vidx[32][8];
    __shared__ float rowm[16];

    const int strip = blockIdx.x;
    const int n     = blockIdx.y;
    const int lane  = threadIdx.x;
    const int hl    = lane >> 4;
    const int ln    = lane & 15;
    const int prow  = strip * 16 + ln;

    float px = 0.f, py = 0.f, pz = 0.f;
    if (prow < P) {
        const float* pp = pred + ((size_t)n * P + prow) * 3;
        px = pp[0]; py = pp[1]; pz = pp[2];
    }
    if (hl == 0) x2s[ln] = px * px + py * py + pz * pz;
    __syncthreads();

    v2f a;
    a.x = hl ? pz : px;
    a.y = hl ? 1.0f : py;

    float runv[8]; int runt[8];
#pragma unroll
    for (int r = 0; r < 8; ++r) { runv[r] = 3.4e38f; runt[r] = 0; }

    const float* gbase = gt + ((size_t)n * PGT + ln) * 3;
    if (n == 0) {
        // track argmin (only batch 0's indices are consumed)
#pragma unroll 2
        for (int t = 0; t < PGT / 16; ++t) {
            const float* gq = gbase + (size_t)t * 48;
            const float gx = gq[0], gy = gq[1], gz = gq[2];
            const float g2 = gx * gx + gy * gy + gz * gz;
            v2f b;
            b.x = -2.f * (hl ? gz : gx);
            b.y = hl ? g2 : (-2.f * gy);
            v8f c = {};                    // inline 0 C operand
            c = wmma_dist(a, b, c);
#pragma unroll
            for (int r = 0; r < 8; ++r) {
                if (c[r] < runv[r]) { runv[r] = c[r]; runt[r] = t; }
            }
        }
    } else {
        // value-only min: 8 v_min_num_f32 per tile
#pragma unroll 2
        for (int t = 0; t < PGT / 16; ++t) {
            const float* gq = gbase + (size_t)t * 48;
            const float gx = gq[0], gy = gq[1], gz = gq[2];
            const float g2 = gx * gx + gy * gy + gz * gz;
            v2f b;
            b.x = -2.f * (hl ? gz : gx);
            b.y = hl ? g2 : (-2.f * gy);
            v8f c = {};
            c = wmma_dist(a, b, c);
#pragma unroll
            for (int r = 0; r < 8; ++r) runv[r] = fmin1(runv[r], c[r]);
        }
    }

#pragma unroll
    for (int r = 0; r < 8; ++r) {
        vmin[lane][r] = runv[r];
        vidx[lane][r] = runt[r] * 16 + ln;
    }
    __syncthreads();
    if (lane < 16) {
        const int rr = lane;               // output pred row within strip
        const int h = rr >> 3, r = rr & 7; // source half / VGPR slot
        float best = 3.4e38f; int bidx = 0;
        for (int l2 = 0; l2 < 16; ++l2) {
            const float v = vmin[h * 16 + l2][r];
            if (v < best) { best = v; bidx = vidx[h * 16 + l2][r]; }
        }
        const int myrow = strip * 16 + rr;
        const bool valid = myrow < P;
        rowm[rr] = valid ? (best + x2s[rr]) : 0.f;   // add row-constant |x|^2
        if (valid && n == 0) idx0[myrow] = bidx;
    }
    __syncthreads();
    if (lane == 0) {
        float s = 0.f;
        for (int k = 0; k < 16; ++k) s += rowm[k];
        atomicAdd(accum, s);
    }
}

// ---------------------------------------------------------------------------
// Chamfer gt->pred: one wave per (gt strip of 16, batch). Unchecked main loop
// over full pred tiles; single checked tail tile (padded cols get huge
// coordinates so they never win the min). |g|^2 added after the reduction.
// ---------------------------------------------------------------------------
__global__ __launch_bounds__(32) void chamfer_gp(
        const float* __restrict__ pred, const float* __restrict__ gt,
        int P, float* __restrict__ accum) {
    __shared__ float g2s[16];
    __shared__ float vmin[32][8];
    __shared__ float rowm[16];

    const int strip = blockIdx.x;       // 0..624
    const int n     = blockIdx.y;
    const int lane  = threadIdx.x;
    const int hl    = lane >> 4;
    const int ln    = lane & 15;
    const int grow  = strip * 16 + ln;

    const float* gq = gt + ((size_t)n * PGT + grow) * 3;
    const float gx = gq[0], gy = gq[1], gz = gq[2];
    if (hl == 0) g2s[ln] = gx * gx + gy * gy + gz * gz;
    __syncthreads();

    v2f a;
    a.x = hl ? gz : gx;
    a.y = hl ? 1.0f : gy;

    float runv[8];
#pragma unroll
    for (int r = 0; r < 8; ++r) runv[r] = 3.4e38f;

    const int nfull = P / 16;           // full tiles, no bounds check
    const float* pbase = pred + ((size_t)n * P + ln) * 3;
#pragma unroll 2
    for (int t = 0; t < nfull; ++t) {
        const float* pp = pbase + (size_t)t * 48;
        const float px = pp[0], py = pp[1], pz = pp[2];
        const float p2 = px * px + py * py + pz * pz;
        v2f b;
        b.x = -2.f * (hl ? pz : px);
        b.y = hl ? p2 : (-2.f * py);
        v8f c = {};
        c = wmma_dist(a, b, c);
#pragma unroll
        for (int r = 0; r < 8; ++r) runv[r] = fmin1(runv[r], c[r]);
    }
    if (nfull * 16 < P) {               // one tail tile with padding
        const int pc = nfull * 16 + ln;
        float px = 1e15f, py = 1e15f, pz = 1e15f;   // pad -> |y|^2 ~ 3e30
        if (pc < P) {
            const float* pp = pbase + (size_t)nfull * 48;
            px = pp[0]; py = pp[1]; pz = pp[2];
        }
        const float p2 = px * px + py * py + pz * pz;
        v2f b;
        b.x = -2.f * (hl ? pz : px);
        b.y = hl ? p2 : (-2.f * py);
        v8f c = {};
        c = wmma_dist(a, b, c);
#pragma unroll
        for (int r = 0; r < 8; ++r) runv[r] = fmin1(runv[r], c[r]);
    }

#pragma unroll
    for (int r = 0; r < 8; ++r) vmin[lane][r] = runv[r];
    __syncthreads();
    if (lane < 16) {
        const int h = lane >> 3, r = lane & 7;
        float best = 3.4e38f;
        for (int l2 = 0; l2 < 16; ++l2) best = fmin1(best, vmin[h * 16 + l2][r]);
        rowm[lane] = best + g2s[lane];              // add row-constant |g|^2
    }
    __syncthreads();
    if (lane == 0) {
        float s = 0.f;
        for (int k = 0; k < 16; ++k) s += rowm[k];
        atomicAdd(accum, s);
    }
}

// ---------------------------------------------------------------------------
// Edge + normal losses (consumes idx_pg0 from chamfer_pg, batch 0)
// ---------------------------------------------------------------------------
__global__ void edge_normal(const float* __restrict__ pred, const int* __restrict__ edges,
                            const float* __restrict__ gtn, const int* __restrict__ idx0,
                            int P, int E, float* accE, float* accN) {
    __shared__ float sE[256], sN[256];
    const int tid = blockIdx.x * blockDim.x + threadIdx.x;
    const int stride = gridDim.x * blockDim.x;
    float eacc = 0.f, nacc = 0.f;
    const int total = NB * E;
    for (int i = tid; i < total; i += stride) {
        const int n = i / E, e = i - n * E;
        const int va = edges[2 * e], vb = edges[2 * e + 1];
        const float* pa = pred + ((size_t)n * P + va) * 3;
        const float* pb = pred + ((size_t)n * P + vb) * 3;
        const float dx = pa[0] - pb[0], dy = pa[1] - pb[1], dz = pa[2] - pb[2];
        const float sq = dx * dx + dy * dy + dz * dz;
        eacc += sq;
        const float inv = 1.f / fmaxf(sqrtf(sq), 1e-12f);
        const int ni = idx0[va];
        const float* nv = gtn + ((size_t)n * PGT + ni) * 3;
        const float nx = nv[0], ny = nv[1], nz = nv[2];
        const float ninv = 1.f / fmaxf(sqrtf(nx * nx + ny * ny + nz * nz), 1e-12f);
        nacc += fabsf((dx * nx + dy * ny + dz * nz) * inv * ninv);
    }
    sE[threadIdx.x] = eacc; sN[threadIdx.x] = nacc;
    __syncthreads();
    for (int s = blockDim.x / 2; s > 0; s >>= 1) {
        if ((int)threadIdx.x < s) {
            sE[threadIdx.x] += sE[threadIdx.x + s];
            sN[threadIdx.x] += sN[threadIdx.x + s];
        }
        __syncthreads();
    }
    if (threadIdx.x == 0) { atomicAdd(accE, sE[0]); atomicAdd(accN, sN[0]); }
}

// ---------------------------------------------------------------------------
// Laplacian + move losses
// ---------------------------------------------------------------------------
__global__ void lap_move(const float* __restrict__ pred, const float* __restrict__ before,
                         const int* __restrict__ lap, int P, float* accL, float* accM) {
    __shared__ float sL[256], sM[256];
    const int tid = blockIdx.x * blockDim.x + threadIdx.x;
    const int stride = gridDim.x * blockDim.x;
    float lacc = 0.f, macc = 0.f;
    const int total = NB * P;
    for (int i = tid; i < total; i += stride) {
        const int n = i / P, p = i - n * P;
        const int* li = lap + (size_t)p * 10;
        const float inv = 1.f / (float)li[9];
        float sb0 = 0, sb1 = 0, sb2 = 0, sp0 = 0, sp1 = 0, sp2 = 0;
#pragma unroll
        for (int k = 0; k < 8; ++k) {
            const int id = li[k];
            if (id >= 0) {
                const float* bb = before + ((size_t)n * P + id) * 3;
                const float* pp = pred + ((size_t)n * P + id) * 3;
                sb0 += bb[0]; sb1 += bb[1]; sb2 += bb[2];
                sp0 += pp[0]; sp1 += pp[1]; sp2 += pp[2];
            }
        }
        const float* bs = before + ((size_t)n * P + p) * 3;
        const float* ps = pred + ((size_t)n * P + p) * 3;
        const float d0 = (bs[0] - sb0 * inv) - (ps[0] - sp0 * inv);
        const float d1 = (bs[1] - sb1 * inv) - (ps[1] - sp1 * inv);
        const float d2 = (bs[2] - sb2 * inv) - (ps[2] - sp2 * inv);
        lacc += d0 * d0 + d1 * d1 + d2 * d2;
        const float m0 = bs[0] - ps[0], m1 = bs[1] - ps[1], m2 = bs[2] - ps[2];
        macc += m0 * m0 + m1 * m1 + m2 * m2;
    }
    sL[threadIdx.x] = lacc; sM[threadIdx.x] = macc;
    __syncthreads();
    for (int s = blockDim.x / 2; s > 0; s >>= 1) {
        if ((int)threadIdx.x < s) {
            sL[threadIdx.x] += sL[threadIdx.x + s];
            sM[threadIdx.x] += sM[threadIdx.x + s];
        }
        __syncthreads();
    }
    if (threadIdx.x == 0) {
        atomicAdd(accL, sL[0]);
        if (accM) atomicAdd(accM, sM[0]);
    }
}

// ---------------------------------------------------------------------------
// Image binary cross-entropy
// ---------------------------------------------------------------------------
__global__ void image_bce(const float* __restrict__ g, const float* __restrict__ r,
                          int total, float* acc) {
    __shared__ float s[256];
    const int tid = blockIdx.x * blockDim.x + threadIdx.x;
    const int stride = gridDim.x * blockDim.x;
    float a = 0.f;
    for (int i = tid; i < total; i += stride) {
        const float gi = g[i], ri = r[i];
        a -= gi * logf(ri) + (1.f - gi) * logf(1.f - ri);
    }
    s[threadIdx.x] = a;
    __syncthreads();
    for (int k = blockDim.x / 2; k > 0; k >>= 1) {
        if ((int)threadIdx.x < k) s[threadIdx.x] += s[threadIdx.x + k];
        __syncthreads();
    }
    if (threadIdx.x == 0) atomicAdd(acc, s[0]);
}

// ---------------------------------------------------------------------------
// Final scalar combine
// ---------------------------------------------------------------------------
__global__ void finalize(const float* acc, float* out) {
    if (threadIdx.x != 0 || blockIdx.x != 0) return;
    const float Pl[3] = {156.f, 618.f, 2466.f};
    const float El[3] = {462.f, 1848.f, 7392.f};
    const float lapc[3] = {0.2f, 1.f, 1.f};
    float chamfer = 0.f, edge = 0.f, normal = 0.f, lap = 0.f;
    for (int i = 0; i < 3; ++i) {
        chamfer += acc[2 * i] / (4.f * Pl[i]) + 0.55f * acc[2 * i + 1] / (4.f * 10000.f);
        edge    += acc[6 + i] / (4.f * El[i]);     // mse*3 over (N,E,3) = sum/(N*E)
        normal  += acc[9 + i] / (4.f * El[i]);
        lap     += lapc[i] * acc[12 + i] / (4.f * Pl[i]);
    }
    const float move = acc[15] / (4.f * 618.f) + acc[16] / (4.f * 2466.f);
    const float image = acc[17] / (4.f * 3.f * 224.f * 224.f);
    out[0] = chamfer + 1.0f * image + 0.5f * lap + 0.1f * move + 0.1f * edge + 0.00016f * normal;
}

// ---------------------------------------------------------------------------
extern "C" void kernel_launch(void* const* d_in, const int* in_sizes, int n_in,
                              void* d_out, int out_size, void* d_ws, size_t ws_size,
                              hipStream_t stream) {
    const float* pred[3]   = {(const float*)d_in[0], (const float*)d_in[4], (const float*)d_in[8]};
    const float* before[3] = {(const float*)d_in[1], (const float*)d_in[5], (const float*)d_in[9]};
    const int*   edges[3]  = {(const int*)d_in[2],  (const int*)d_in[6],  (const int*)d_in[10]};
    const int*   lapi[3]   = {(const int*)d_in[3],  (const int*)d_in[7],  (const int*)d_in[11]};
    const float* gtp  = (const float*)d_in[12];
    const float* gtn  = (const float*)d_in[13];
    const float* gimg = (const float*)d_in[14];
    const float* rimg = (const float*)d_in[15];
    const int P[3] = {156, 618, 2466};
    const int E[3] = {462, 1848, 7392};

    float* acc = (float*)d_ws;                 // 32 float slots
    int* idxbase = (int*)d_ws + 32;
    int* idx0[3] = {idxbase, idxbase + 256, idxbase + 1024};  // padded regions

    zero_acc<<<1, 32, 0, stream>>>(acc);

    for (int i = 0; i < 3; ++i) {
        const int strips = (P[i] + 15) / 16;
        chamfer_pg<<<dim3(strips, NB), 32, 0, stream>>>(pred[i], gtp, P[i], acc + 2 * i, idx0[i]);
        chamfer_gp<<<dim3(PGT / 16, NB), 32, 0, stream>>>(pred[i], gtp, P[i], acc + 2 * i + 1);
    }
    for (int i = 0; i < 3; ++i) {
        const int totE = NB * E[i];
        int blocksE = (totE + 255) / 256; if (blocksE > 512) blocksE = 512;
        edge_normal<<<blocksE, 256, 0, stream>>>(pred[i], edges[i], gtn, idx0[i],
                                                 P[i], E[i], acc + 6 + i, acc + 9 + i);
        const int totP = NB * P[i];
        const int blocksP = (totP + 255) / 256;
        lap_move<<<blocksP, 256, 0, stream>>>(pred[i], before[i], lapi[i], P[i],
                                              acc + 12 + i, (i > 0) ? (acc + 14 + i) : nullptr);
    }
    const int itotal = NB * 3 * 224 * 224;
    image_bce<<<512, 256, 0, stream>>>(gimg, rimg, itotal, acc + 17);
    finalize<<<1, 1, 0, stream>>>(acc, (float*)d_out);
}